// wsi_model_86079734546517
// MI455X (gfx1250) — compile-verified
//
#include <hip/hip_runtime.h>

#define G 4
#define NN 20000
#define EE 320000
#define KK1 12000
#define KK2 7200
#define KK3 3600
#define EPSV 1e-5f

typedef __attribute__((ext_vector_type(2))) float v2f;
typedef __attribute__((ext_vector_type(8))) float v8f;

// ---------------- utility kernels ----------------
__global__ void fill_f32(float* p, float v, int n) {
  int i = blockIdx.x * blockDim.x + threadIdx.x;
  if (i < n) p[i] = v;
}
__global__ void fill_u32(unsigned* p, unsigned v, int n) {
  int i = blockIdx.x * blockDim.x + threadIdx.x;
  if (i < n) p[i] = v;
}

// ---------------- GCN pieces ----------------
__global__ void deg_kernel(const int* __restrict__ dst, const float* __restrict__ w,
                           float* __restrict__ deg, int e) {
  int i = blockIdx.x * blockDim.x + threadIdx.x;
  if (i >= e) return;
  float we = w[i];
  if (we != 0.f) atomicAdd(&deg[dst[i]], we);
}

__global__ void dinv_kernel(const float* __restrict__ deg, float* __restrict__ dinv, int n) {
  int i = blockIdx.x * blockDim.x + threadIdx.x;
  if (i >= n) return;
  float d = deg[i];
  dinv[i] = (d > 0.f) ? rsqrtf(d) : 0.f;
}

// fp32 WMMA GEMM: Y[n,dout] = X[n,din] @ W[din,dout]
// block = 256 threads = 8 waves; wave handles one 16x16 output tile.
// grid = (n/16, dout/128). No divergence before WMMA (EXEC must be all ones).
__global__ void gemm_wmma(const float* __restrict__ X, const float* __restrict__ W,
                          float* __restrict__ Y, int n, int din, int dout) {
  const int wave = threadIdx.x >> 5;
  const int lane = threadIdx.x & 31;
  const int m    = lane & 15;
  const int hi   = lane >> 4;            // half-wave select (K pair / M+8)
  const int tm   = blockIdx.x * 16;
  const int tn   = (blockIdx.y * 8 + wave) * 16;

  const float* xrow = X + (size_t)(tm + m) * din;
  v8f acc = {};
  for (int k0 = 0; k0 < din; k0 += 4) {
    // A 16x4 fp32: VGPR0 = A[m, hi?2:0], VGPR1 = A[m, hi?3:1] (consecutive -> float2)
    const float2 av = *(const float2*)(xrow + k0 + 2 * hi);
    v2f a, b;
    a.x = av.x; a.y = av.y;
    // B 4x16 fp32 (mirror layout): VGPR0 = B[hi?2:0, m], VGPR1 = B[hi?3:1, m]
    b.x = W[(size_t)(k0 + 2 * hi)     * dout + tn + m];
    b.y = W[(size_t)(k0 + 2 * hi + 1) * dout + tn + m];
    acc = __builtin_amdgcn_wmma_f32_16x16x4_f32(false, a, false, b, (short)0, acc,
                                                false, false);
  }
  // C/D: VGPR r -> row (tm + r + 8*hi), col tn+m
#pragma unroll
  for (int r = 0; r < 8; ++r)
    Y[(size_t)(tm + 8 * hi + r) * dout + tn + m] = acc[r];
}

// scatter: out[dst] += (w * dinv[src] * dinv[dst]) * h[src], 4 features per thread
__global__ void gcn_scatter(const float* __restrict__ h, const int* __restrict__ src,
                            const int* __restrict__ dst, const float* __restrict__ w,
                            const float* __restrict__ dinv, float* __restrict__ out,
                            int e, int D_) {
  int idx = blockIdx.x * blockDim.x + threadIdx.x;
  int chunks = D_ >> 2;
  if (idx >= e * chunks) return;
  int ei = idx / chunks;
  int c  = (idx - ei * chunks) << 2;
  float we = w[ei];
  if (we == 0.f) return;
  int s = src[ei], d = dst[ei];
  float norm = we * dinv[s] * dinv[d];
  const float4 hv = *(const float4*)(h + (size_t)s * D_ + c);
  float* o = out + (size_t)d * D_ + c;
  atomicAdd(o + 0, norm * hv.x);
  atomicAdd(o + 1, norm * hv.y);
  atomicAdd(o + 2, norm * hv.z);
  atomicAdd(o + 3, norm * hv.w);
}

// ---------------- BatchNorm ----------------
__global__ void bn_stats(const float* __restrict__ h, float* __restrict__ mean,
                         float* __restrict__ rstd, int n, int D_) {
  int j = blockIdx.x;
  float s = 0.f, s2 = 0.f;
  for (int i = threadIdx.x; i < n; i += blockDim.x) {
    float v = h[(size_t)i * D_ + j];
    s += v; s2 += v * v;
  }
  __shared__ float sa[256], sb[256];
  sa[threadIdx.x] = s; sb[threadIdx.x] = s2;
  __syncthreads();
  for (int st = 128; st; st >>= 1) {
    if (threadIdx.x < st) { sa[threadIdx.x] += sa[threadIdx.x + st];
                            sb[threadIdx.x] += sb[threadIdx.x + st]; }
    __syncthreads();
  }
  if (threadIdx.x == 0) {
    float mu = sa[0] / (float)n;
    float var = sb[0] / (float)n - mu * mu;
    mean[j] = mu;
    rstd[j] = rsqrtf(var + EPSV);
  }
}

__global__ void bn_apply_relu(float* __restrict__ h, const float* __restrict__ mean,
                              const float* __restrict__ rstd, const float* __restrict__ gam,
                              const float* __restrict__ bet, int total, int D_) {
  int idx = blockIdx.x * blockDim.x + threadIdx.x;
  if (idx >= total) return;
  int j = idx % D_;
  float v = (h[idx] - mean[j]) * rstd[j] * gam[j] + bet[j];
  h[idx] = v > 0.f ? v : 0.f;
}

// ---------------- SAGPool ----------------
// per node: r = x.w_rel, t = x.w_root  (one wave per node)
__global__ void pool_score_lin(const float* __restrict__ h, const float* __restrict__ wrel,
                               const float* __restrict__ wroot, float* __restrict__ r,
                               float* __restrict__ t, int n, int D_) {
  int gid  = blockIdx.x * blockDim.x + threadIdx.x;
  int node = gid >> 5;
  int lane = gid & 31;
  if (node >= n) return;
  float sr = 0.f, st = 0.f;
  const float* row = h + (size_t)node * D_;
  for (int d = lane; d < D_; d += 32) {
    float v = row[d];
    sr += v * wrel[d];
    st += v * wroot[d];
  }
  for (int off = 16; off; off >>= 1) {
    sr += __shfl_down(sr, off, 32);
    st += __shfl_down(st, off, 32);
  }
  if (lane == 0) { r[node] = sr; t[node] = st; }
}

__global__ void scatter_score(const float* __restrict__ r, const int* __restrict__ src,
                              const int* __restrict__ dst, const float* __restrict__ w,
                              float* __restrict__ aggs, int e) {
  int i = blockIdx.x * blockDim.x + threadIdx.x;
  if (i >= e) return;
  float we = w[i];
  if (we == 0.f) return;
  atomicAdd(&aggs[dst[i]], we * r[src[i]]);
}

__global__ void score_tanh(const float* __restrict__ aggs, const float* __restrict__ t,
                           const float* __restrict__ brel, float* __restrict__ score, int n) {
  int i = blockIdx.x * blockDim.x + threadIdx.x;
  if (i >= n) return;
  score[i] = tanhf(aggs[i] + brel[0] + t[i]);
}

__device__ __forceinline__ unsigned mapf(float f) {
  unsigned u = __float_as_uint(f);
  return (u & 0x80000000u) ? ~u : (u | 0x80000000u);   // monotone: bigger float -> bigger uint
}

__global__ void topk_init(unsigned* state, int k) {
  if (threadIdx.x == 0) { state[0] = 0u; state[1] = (unsigned)k; state[2] = 0u; state[3] = 0u; }
}

__global__ void topk_hist(const float* __restrict__ score, int n, unsigned* __restrict__ hist,
                          const unsigned* __restrict__ state, int round) {
  __shared__ unsigned lh[256];
  if (threadIdx.x < 256) lh[threadIdx.x] = 0u;
  __syncthreads();
  int shift = 24 - 8 * round;
  unsigned prefix = state[0];
  for (int i = blockIdx.x * blockDim.x + threadIdx.x; i < n; i += gridDim.x * blockDim.x) {
    unsigned u = mapf(score[i]);
    bool match = (round == 0) || ((u >> (shift + 8)) == (prefix >> (shift + 8)));
    if (match) atomicAdd(&lh[(u >> shift) & 255u], 1u);
  }
  __syncthreads();
  if (threadIdx.x < 256) atomicAdd(&hist[threadIdx.x], lh[threadIdx.x]);
}

__global__ void topk_scan(const unsigned* __restrict__ hist, unsigned* __restrict__ state,
                          int round) {
  if (threadIdx.x != 0 || blockIdx.x != 0) return;
  unsigned prefix = state[0];
  int krem = (int)state[1];
  int shift = 24 - 8 * round;
  int b = 255;
  for (; b >= 0; --b) {
    int c = (int)hist[b];
    if (c >= krem) break;
    krem -= c;
  }
  if (b < 0) b = 0;
  state[0] = prefix | ((unsigned)b << shift);
  state[1] = (unsigned)krem;
  if (round == 3) { state[2] = 0u; state[3] = 0u; }      // reset select counters
}

__global__ void topk_select(const float* __restrict__ score, int n,
                            unsigned* __restrict__ state, int* __restrict__ mapping,
                            int* __restrict__ selnode) {
  int i = blockIdx.x * blockDim.x + threadIdx.x;
  if (i >= n) return;
  unsigned T = state[0];
  unsigned eq_needed = state[1];
  unsigned u = mapf(score[i]);
  int slot = -1;
  if (u > T) {
    slot = (int)atomicAdd(&state[2], 1u);
  } else if (u == T) {
    unsigned e = atomicAdd(&state[3], 1u);
    if (e < eq_needed) slot = (int)atomicAdd(&state[2], 1u);
  }
  mapping[i] = slot;
  if (slot >= 0) selnode[slot] = i;
}

__global__ void gather_scale(const float* __restrict__ hin, const int* __restrict__ selnode,
                             const float* __restrict__ score, float* __restrict__ hout,
                             int k, int D_) {
  int idx = blockIdx.x * blockDim.x + threadIdx.x;
  int chunks = D_ >> 2;
  if (idx >= k * chunks) return;
  int nw = idx / chunks;
  int c  = (idx - nw * chunks) << 2;
  int i  = selnode[nw];
  float s = score[i];
  float4 v = *(const float4*)(hin + (size_t)i * D_ + c);
  float4 o = make_float4(v.x * s, v.y * s, v.z * s, v.w * s);
  *(float4*)(hout + (size_t)nw * D_ + c) = o;
}

__global__ void reindex_edges(const int* __restrict__ src, const int* __restrict__ dst,
                              const float* __restrict__ w, const int* __restrict__ mapping,
                              int* __restrict__ src2, int* __restrict__ dst2,
                              float* __restrict__ w2, int e) {
  int i = blockIdx.x * blockDim.x + threadIdx.x;
  if (i >= e) return;
  int s = mapping[src[i]];
  int d = mapping[dst[i]];
  bool valid = (s >= 0) && (d >= 0) && (w[i] > 0.f);
  src2[i] = s > 0 ? s : 0;
  dst2[i] = d > 0 ? d : 0;
  w2[i]   = valid ? 1.f : 0.f;
}

// ---------------- pooling epilogue + head ----------------
__global__ void maxpool_bias(const float* __restrict__ h, const float* __restrict__ bias,
                             float* __restrict__ feat, int n, int D_) {
  int j = blockIdx.x;
  float m = -3.402823466e38f;
  for (int i = threadIdx.x; i < n; i += blockDim.x)
    m = fmaxf(m, h[(size_t)i * D_ + j]);
  __shared__ float sh[256];
  sh[threadIdx.x] = m;
  __syncthreads();
  for (int s = 128; s; s >>= 1) {
    if (threadIdx.x < s) sh[threadIdx.x] = fmaxf(sh[threadIdx.x], sh[threadIdx.x + s]);
    __syncthreads();
  }
  if (threadIdx.x == 0) feat[j] = sh[0] + bias[j];
}

__global__ void head_kernel(const float* __restrict__ feat, const float* __restrict__ lin_w,
                            const float* __restrict__ lin_b, const float* __restrict__ lin2_w,
                            const float* __restrict__ lin2_b, float* __restrict__ out) {
  __shared__ float x1[256];
  __shared__ float red[256];
  int j = threadIdx.x;
  float v = 0.25f * (feat[j] + feat[256 + j] + feat[512 + j] + feat[768 + j]);
  x1[j] = v;
  out[1 + j] = v;                      // tuple order: (out, x1) -> x1 at [1..256]
  __syncthreads();
  float hsum = lin_b[j];
  for (int i = 0; i < 256; ++i) hsum += x1[i] * lin_w[i * 256 + j];
  float hr = hsum > 0.f ? hsum : 0.f;
  red[j] = hr * lin2_w[j];
  __syncthreads();
  for (int s = 128; s; s >>= 1) {
    if (j < s) red[j] += red[j + s];
    __syncthreads();
  }
  if (j == 0) out[0] = red[0] + lin2_b[0];
}

// ---------------- host orchestration ----------------
static inline int gridFor(long n) { return (int)((n + 255) / 256); }

struct Bufs {
  float *deg, *dinv, *r, *t, *aggs, *score, *mean, *rstd, *feat, *wOnes;
  int *mapping, *selnode;
  unsigned *hist, *state;
};

static void gcn_level(hipStream_t s, const float* xin, float* tmp, float* xout,
                      const int* src, const int* dst, const float* w,
                      const float* W, int n, int din, int dout, const Bufs& B) {
  fill_f32<<<gridFor(n), 256, 0, s>>>(B.deg, 0.f, n);
  deg_kernel<<<gridFor(EE), 256, 0, s>>>(dst, w, B.deg, EE);
  dinv_kernel<<<gridFor(n), 256, 0, s>>>(B.deg, B.dinv, n);
  dim3 gg(n / 16, dout / 128);
  gemm_wmma<<<gg, 256, 0, s>>>(xin, W, tmp, n, din, dout);
  fill_f32<<<gridFor((long)n * dout), 256, 0, s>>>(xout, 0.f, n * dout);
  long total = (long)EE * (dout / 4);
  gcn_scatter<<<gridFor(total), 256, 0, s>>>(tmp, src, dst, w, B.dinv, xout, EE, dout);
}

static void bn_level(hipStream_t s, float* h, const float* g, const float* be,
                     int n, int d, const Bufs& B) {
  bn_stats<<<d, 256, 0, s>>>(h, B.mean, B.rstd, n, d);
  bn_apply_relu<<<gridFor((long)n * d), 256, 0, s>>>(h, B.mean, B.rstd, g, be, n * d, d);
}

static void pool_level(hipStream_t s, const float* xin, float* xout,
                       const int* src, const int* dst, const float* w,
                       int* src2, int* dst2, float* w2,
                       const float* wrel, const float* brel, const float* wroot,
                       int n, int d, int k, const Bufs& B) {
  pool_score_lin<<<gridFor((long)n * 32), 256, 0, s>>>(xin, wrel, wroot, B.r, B.t, n, d);
  fill_f32<<<gridFor(n), 256, 0, s>>>(B.aggs, 0.f, n);
  scatter_score<<<gridFor(EE), 256, 0, s>>>(B.r, src, dst, w, B.aggs, EE);
  score_tanh<<<gridFor(n), 256, 0, s>>>(B.aggs, B.t, brel, B.score, n);
  topk_init<<<1, 32, 0, s>>>(B.state, k);
  for (int round = 0; round < 4; ++round) {
    fill_u32<<<1, 256, 0, s>>>(B.hist, 0u, 256);
    topk_hist<<<64, 256, 0, s>>>(B.score, n, B.hist, B.state, round);
    topk_scan<<<1, 32, 0, s>>>(B.hist, B.state, round);
  }
  topk_select<<<gridFor(n), 256, 0, s>>>(B.score, n, B.state, B.mapping, B.selnode);
  gather_scale<<<gridFor((long)k * (d / 4)), 256, 0, s>>>(xin, B.selnode, B.score, xout, k, d);
  reindex_edges<<<gridFor(EE), 256, 0, s>>>(src, dst, w, B.mapping, src2, dst2, w2, EE);
}

extern "C" void kernel_launch(void* const* d_in, const int* in_sizes, int n_in,
                              void* d_out, int out_size, void* d_ws, size_t ws_size,
                              hipStream_t stream) {
  (void)in_sizes; (void)n_in; (void)out_size; (void)ws_size;
  const float* x          = (const float*)d_in[0];
  const int*   edge_index = (const int*)d_in[1];
  const float* W1 = (const float*)d_in[2];
  const float* W2 = (const float*)d_in[4];
  const float* W3 = (const float*)d_in[6];
  const float* W4 = (const float*)d_in[8];
  const float* b4 = (const float*)d_in[9];
  const float* g1 = (const float*)d_in[10]; const float* be1 = (const float*)d_in[11];
  const float* g2 = (const float*)d_in[12]; const float* be2 = (const float*)d_in[13];
  const float* g3 = (const float*)d_in[14]; const float* be3 = (const float*)d_in[15];
  const float* p1_wrel = (const float*)d_in[16]; const float* p1_brel = (const float*)d_in[17];
  const float* p1_wroot = (const float*)d_in[18];
  const float* p2_wrel = (const float*)d_in[19]; const float* p2_brel = (const float*)d_in[20];
  const float* p2_wroot = (const float*)d_in[21];
  const float* p3_wrel = (const float*)d_in[22]; const float* p3_brel = (const float*)d_in[23];
  const float* p3_wroot = (const float*)d_in[24];
  const float* lin_w  = (const float*)d_in[25]; const float* lin_b  = (const float*)d_in[26];
  const float* lin2_w = (const float*)d_in[27]; const float* lin2_b = (const float*)d_in[28];
  float* out = (float*)d_out;

  // ---- workspace layout ----
  char* ws = (char*)d_ws;
  size_t off = 0;
  auto alloc = [&](size_t bytes) -> void* {
    void* p = ws + off;
    off += (bytes + 255) & ~(size_t)255;
    return p;
  };
  float* hA = (float*)alloc((size_t)NN * 512 * 4);
  float* hB = (float*)alloc((size_t)NN * 512 * 4);
  Bufs B;
  B.deg   = (float*)alloc((size_t)NN * 4);
  B.dinv  = (float*)alloc((size_t)NN * 4);
  B.r     = (float*)alloc((size_t)NN * 4);
  B.t     = (float*)alloc((size_t)NN * 4);
  B.aggs  = (float*)alloc((size_t)NN * 4);
  B.score = (float*)alloc((size_t)NN * 4);
  B.mapping = (int*)alloc((size_t)NN * 4);
  B.selnode = (int*)alloc((size_t)NN * 4);
  int*   srcA = (int*)alloc((size_t)EE * 4);
  int*   dstA = (int*)alloc((size_t)EE * 4);
  float* wA   = (float*)alloc((size_t)EE * 4);
  int*   srcB = (int*)alloc((size_t)EE * 4);
  int*   dstB = (int*)alloc((size_t)EE * 4);
  float* wB   = (float*)alloc((size_t)EE * 4);
  B.wOnes = (float*)alloc((size_t)EE * 4);
  B.hist  = (unsigned*)alloc(256 * 4);
  B.state = (unsigned*)alloc(8 * 4);
  B.mean  = (float*)alloc(512 * 4);
  B.rstd  = (float*)alloc(512 * 4);
  B.feat  = (float*)alloc((size_t)G * 256 * 4);

  fill_f32<<<gridFor(EE), 256, 0, stream>>>(B.wOnes, 1.f, EE);

  for (int g = 0; g < G; ++g) {
    const float* x_g  = x + (size_t)g * NN * 512;
    const int*   src0 = edge_index + (size_t)g * 2 * EE;
    const int*   dst0 = src0 + EE;

    // ---- level 1: GCN(512->512) + BN + ReLU + SAGPool(N -> K1)
    gcn_level(stream, x_g, hB, hA, src0, dst0, B.wOnes, W1, NN, 512, 512, B);
    bn_level(stream, hA, g1, be1, NN, 512, B);
    pool_level(stream, hA, hB, src0, dst0, B.wOnes, srcA, dstA, wA,
               p1_wrel, p1_brel, p1_wroot, NN, 512, KK1, B);
    float* cur = hB; float* other = hA;

    // ---- level 2: GCN(512->512) + BN + ReLU + SAGPool(K1 -> K2)
    gcn_level(stream, cur, other, cur, srcA, dstA, wA, W2, KK1, 512, 512, B);
    bn_level(stream, cur, g2, be2, KK1, 512, B);
    pool_level(stream, cur, other, srcA, dstA, wA, srcB, dstB, wB,
               p2_wrel, p2_brel, p2_wroot, KK1, 512, KK2, B);
    { float* t0 = cur; cur = other; other = t0; }

    // ---- level 3: GCN(512->256) + BN + ReLU + SAGPool(K2 -> K3)
    gcn_level(stream, cur, other, cur, srcB, dstB, wB, W3, KK2, 512, 256, B);
    bn_level(stream, cur, g3, be3, KK2, 256, B);
    pool_level(stream, cur, other, srcB, dstB, wB, srcA, dstA, wA,
               p3_wrel, p3_brel, p3_wroot, KK2, 256, KK3, B);
    { float* t0 = cur; cur = other; other = t0; }

    // ---- level 4: GCN(256->256) + bias-folded global max pool
    gcn_level(stream, cur, other, cur, srcA, dstA, wA, W4, KK3, 256, 256, B);
    maxpool_bias<<<256, 256, 0, stream>>>(cur, b4, B.feat + (size_t)g * 256, KK3, 256);
  }

  head_kernel<<<1, 256, 0, stream>>>(B.feat, lin_w, lin_b, lin2_w, lin2_b, out);
}